// Threshold_weights6_52699248721951
// MI455X (gfx1250) — compile-verified
//
#include <hip/hip_runtime.h>
#include <cstdint>
#include <math.h>

// ---------------------------------------------------------------------------
// Threshold_weights6: per-row top2 margins over 7 heads -> 7-way softmax,
// plus global max over heads 0..5.  Bandwidth-bound (~459 MB @ 23.3 TB/s
// => ~20 us floor).  Data path: CDNA5 async global->LDS staging
// (global_load_async_to_lds_b128 / s_wait_asynccnt) + wave32 shuffle
// reductions.  No matrix structure in this workload -> WMMA not applicable.
// ---------------------------------------------------------------------------

namespace {

constexpr int kHeads = 7;
constexpr int kMaxC  = 1000;   // C == 1000 in this problem; LDS staging sized for it

// CDNA5 async copy: per-lane 16B global -> LDS, tracked by ASYNCcnt.
__device__ __forceinline__ void async_f4_to_lds(uint32_t lds_addr, const float* gaddr) {
  asm volatile("global_load_async_to_lds_b128 %0, %1, off"
               :: "v"(lds_addr), "v"(gaddr)
               : "memory");
}

__device__ __forceinline__ void wait_async_zero() {
  asm volatile("s_wait_asynccnt 0" ::: "memory");
}

} // namespace

__global__ __launch_bounds__(kHeads * 32)
void margins_softmax_kernel(const float* __restrict__ h0, const float* __restrict__ h1,
                            const float* __restrict__ h2, const float* __restrict__ h3,
                            const float* __restrict__ h4, const float* __restrict__ h5,
                            const float* __restrict__ h6, const int* __restrict__ targets,
                            float* __restrict__ out_thresh,     // d_out + 1, (N,7) row-major
                            float* __restrict__ row_head_max,   // ws, N floats
                            int C)
{
  __shared__ __align__(16) float s_buf[kHeads * kMaxC];
  __shared__ float s_margin[kHeads];
  __shared__ float s_rmax[kHeads];

  const int row  = blockIdx.x;
  const int lane = threadIdx.x & 31;
  // Wave id == head id.  threadIdx.x>>5 is wave-uniform by construction on
  // wave32; readfirstlane makes that visible to the compiler -> scalar
  // branches instead of a divergent v_cmpx/exec-mask cascade.
  const int w = __builtin_amdgcn_readfirstlane(threadIdx.x >> 5);

  const float* head;
  switch (w) {
    case 0: head = h0; break;
    case 1: head = h1; break;
    case 2: head = h2; break;
    case 3: head = h3; break;
    case 4: head = h4; break;
    case 5: head = h5; break;
    default: head = h6; break;
  }
  const float* g   = head + (size_t)row * (size_t)C;
  const int    nf4 = C >> 2;           // 250 float4 per row

  float* s_head = &s_buf[w * kMaxC];
  // Generic pointer to LDS: low 32 bits are the workgroup-relative LDS byte
  // address (aperture layout, ISA ch.10.2) -- exactly what VDST expects.
  const uint32_t lbase = (uint32_t)(uintptr_t)s_head;

  // Stage this wave's row into LDS via the async data path: queue all 8
  // transfers back-to-back on ASYNCcnt, then a single wait.
  {
    int f = lane;
#pragma unroll
    for (int i = 0; i < 7; ++i, f += 32) {        // 7 full 512B wave-transfers
      async_f4_to_lds(lbase + (uint32_t)f * 16u, g + (size_t)f * 4);
    }
    if (f < nf4) {                                 // tail: lanes 0..25
      async_f4_to_lds(lbase + (uint32_t)f * 16u, g + (size_t)f * 4);
    }
  }
  wait_async_zero();   // per-wave ASYNCcnt: our whole row is now resident

  // Per-lane top-2 over a disjoint strided slice (branchless min/max).
  float m1 = -INFINITY, m2 = -INFINITY;
  const float4* s4 = reinterpret_cast<const float4*>(s_head);
  {
    int f = lane;
#pragma unroll
    for (int i = 0; i < 8; ++i, f += 32) {
      if (i == 7 && f >= nf4) break;
      float4 v = s4[f];
      float t;
      t = fminf(v.x, m1); m1 = fmaxf(v.x, m1); m2 = fmaxf(t, m2);
      t = fminf(v.y, m1); m1 = fmaxf(v.y, m1); m2 = fmaxf(t, m2);
      t = fminf(v.z, m1); m1 = fmaxf(v.z, m1); m2 = fmaxf(t, m2);
      t = fminf(v.w, m1); m1 = fmaxf(v.w, m1); m2 = fmaxf(t, m2);
    }
  }

  // Wave32 butterfly merge of disjoint (top1, top2) pairs:
  // top1 = max(a1,b1); top2 = max(min(a1,b1), max(a2,b2)).
  for (int off = 16; off > 0; off >>= 1) {
    float o1 = __shfl_xor(m1, off, 32);
    float o2 = __shfl_xor(m2, off, 32);
    float hi = fmaxf(m1, o1);
    float lo = fmaxf(fminf(m1, o1), fmaxf(m2, o2));
    m1 = hi; m2 = lo;
  }

  if (lane == 0) {
    const float tval = s_head[targets[row]];
    s_margin[w] = (tval == m1) ? (m1 - m2) : 0.0f;
    s_rmax[w]   = m1;
  }
  __syncthreads();

  if (threadIdx.x == 0) {
    // Stable 7-way softmax of margins / TEMPERATURE (T = 2.0).
    float mm[kHeads], e[kHeads];
    float mx = -INFINITY;
#pragma unroll
    for (int j = 0; j < kHeads; ++j) { mm[j] = s_margin[j] * 0.5f; mx = fmaxf(mx, mm[j]); }
    float sum = 0.0f;
#pragma unroll
    for (int j = 0; j < kHeads; ++j) { e[j] = expf(mm[j] - mx); sum += e[j]; }
    const float inv = 1.0f / sum;
    float* o = out_thresh + (size_t)row * kHeads;
#pragma unroll
    for (int j = 0; j < kHeads; ++j) o[j] = e[j] * inv;

    // max_preds contribution: max over heads 0..5 of this row (row max == top1).
    float bm = s_rmax[0];
#pragma unroll
    for (int j = 1; j < 6; ++j) bm = fmaxf(bm, s_rmax[j]);
    row_head_max[row] = bm;
  }
}

__global__ __launch_bounds__(1024)
void max_reduce_kernel(const float* __restrict__ in, float* __restrict__ out0, int n)
{
  __shared__ float s[32];
  float m = -INFINITY;
  for (int i = threadIdx.x; i < n; i += 1024) m = fmaxf(m, in[i]);
  for (int off = 16; off > 0; off >>= 1) m = fmaxf(m, __shfl_xor(m, off, 32));
  const int w = threadIdx.x >> 5, lane = threadIdx.x & 31;
  if (lane == 0) s[w] = m;
  __syncthreads();
  if (w == 0) {
    float v = s[lane];
    for (int off = 16; off > 0; off >>= 1) v = fmaxf(v, __shfl_xor(v, off, 32));
    if (lane == 0) *out0 = v;
  }
}

extern "C" void kernel_launch(void* const* d_in, const int* in_sizes, int n_in,
                              void* d_out, int out_size, void* d_ws, size_t ws_size,
                              hipStream_t stream) {
  const float* h0 = (const float*)d_in[0];
  const float* h1 = (const float*)d_in[1];
  const float* h2 = (const float*)d_in[2];
  const float* h3 = (const float*)d_in[3];
  const float* h4 = (const float*)d_in[4];
  const float* h5 = (const float*)d_in[5];
  const float* h6 = (const float*)d_in[6];          // mimic
  const int*   targets = (const int*)d_in[7];
  // d_in[8] = n_test (== N, shapes are static)

  const int N = in_sizes[7];                        // 16384
  const int C = in_sizes[0] / N;                    // 1000

  float* out = (float*)d_out;                       // out[0] = max_preds, out+1 = (N,7)
  float* ws  = (float*)d_ws;                        // N floats of per-row head maxes

  margins_softmax_kernel<<<N, kHeads * 32, 0, stream>>>(
      h0, h1, h2, h3, h4, h5, h6, targets, out + 1, ws, C);
  max_reduce_kernel<<<1, 1024, 0, stream>>>(ws, out, N);
}